// SpGraphAttentionLayer_53807350284813
// MI455X (gfx1250) — compile-verified
//
#include <hip/hip_runtime.h>
#include <hip/hip_bf16.h>

typedef float v2f __attribute__((ext_vector_type(2)));
typedef float v8f __attribute__((ext_vector_type(8)));

#define N_NODES 50000
#define IN_F    100
#define OUT_F   100
#define NREL    100
#define N_EDGES 500000
#define DPAD    304          // padded concat-feature depth (branch1 D=300, branch2 D=304)
#define OUTPAD  112          // 100 padded to 7 * 16
#define TILE_E  64
#define THREADS 256
#define ALPHA   0.2f

// ---------------------------------------------------------------------------
// Prep: write zero-padded weight matrices Wpad[OUTPAD][DPAD] for both branches
// so the GEMM inner loop needs no bounds checks on B loads.
// ---------------------------------------------------------------------------
__global__ __launch_bounds__(THREADS)
void gat_prep_w_kernel(const float* __restrict__ W1,   // [100, 300]
                       const float* __restrict__ W2,   // [100, 304]
                       float* __restrict__ Wpad1,      // [112, 304]
                       float* __restrict__ Wpad2)      // [112, 304]
{
    const int idx = blockIdx.x * blockDim.x + threadIdx.x;
    const int PER = OUTPAD * DPAD;                     // 34048
    if (idx >= 2 * PER) return;
    const int b   = (idx < PER) ? 0 : 1;
    const int rem = idx - b * PER;
    const int n   = rem / DPAD;
    const int k   = rem - n * DPAD;
    if (b == 0) {
        Wpad1[rem] = (n < OUT_F && k < 300) ? W1[n * 300 + k] : 0.0f;
    } else {
        Wpad2[rem] = (n < OUT_F && k < 304) ? W2[n * 304 + k] : 0.0f;
    }
}

// ---------------------------------------------------------------------------
// Per-edge GEMM + attention kernel.
//   eh[e] = [ x[src] | x[dst] | eembed[e] | extra(4, optional) ]   (D = 300/304)
//   em    = eh @ W.T          [TILE_E x 100] via V_WMMA_F32_16X16X4_F32
//   ee    = exp(-leakyrelu(em . w2))
//   atomic: rowsum[src] += ee ; h[src,:] += ee * em[e,:]
// ---------------------------------------------------------------------------
__global__ __launch_bounds__(THREADS)
void gat_edge_kernel(const float* __restrict__ x,       // [N, 100]
                     const int*   __restrict__ edge,    // [2, E]
                     const float* __restrict__ eembed,  // [E, 100]
                     const float* __restrict__ eother,  // [4, E] or nullptr
                     const float* __restrict__ Wpad,    // [112, 304] zero-padded
                     const float* __restrict__ w2,      // [100]
                     int extra,
                     float* __restrict__ h,             // [N, 100] accum
                     float* __restrict__ rowsum)        // [N] accum
{
    __shared__ float sh_eh[TILE_E][DPAD];    // 77824 B
    __shared__ float sh_em[TILE_E][OUTPAD];  // 28672 B
    __shared__ float sh_w2[OUTPAD];
    __shared__ float sh_ee[TILE_E];
    __shared__ int   sh_src[TILE_E];

    const int tid  = threadIdx.x;
    const int wave = tid >> 5;        // 0..7 (wave32)
    const int lane = tid & 31;

    if (tid < OUTPAD) sh_w2[tid] = (tid < OUT_F) ? w2[tid] : 0.0f;

    // ---- Phase 1: gather eh tile into LDS (4 threads per edge, 76 cols each)
    const int e_base = blockIdx.x * TILE_E;
    const int er = tid & (TILE_E - 1);   // edge row within tile
    const int cg = tid >> 6;             // column group 0..3
    const int e  = e_base + er;
    const bool valid = (e < N_EDGES);
    const int src = valid ? edge[e] : 0;
    const int dst = valid ? edge[N_EDGES + e] : 0;
    if (cg == 0) sh_src[er] = valid ? src : -1;

    #pragma unroll 4
    for (int i = 0; i < DPAD / 4; ++i) {
        const int col = cg * (DPAD / 4) + i;
        float v = 0.0f;
        if (valid) {
            if (col < IN_F)                 v = x[src * IN_F + col];
            else if (col < 2 * IN_F)        v = x[dst * IN_F + (col - IN_F)];
            else if (col < 2 * IN_F + NREL) v = eembed[e * NREL + (col - 2 * IN_F)];
            else if (col < 2 * IN_F + NREL + extra)
                                            v = eother[(col - (2 * IN_F + NREL)) * N_EDGES + e];
        }
        sh_eh[er][col] = v;
    }
    __syncthreads();

    // ---- Phase 2: em = eh @ Wpad.T. Wave w (0..6) owns output subtile w and
    //      keeps 4 edge-subtile accumulators live -> per k-step:
    //      1x global_load_b64 (B) + 4x ds_load_b64 (A) + 4x WMMA.
    const int m    = lane & 15;
    const int koff = (lane < 16) ? 0 : 2;   // ISA half-wave K striping (A & B)
    if (wave < 7) {
        const int ncol = wave * 16 + m;     // B column (output feature, padded)
        const float* __restrict__ wcol = Wpad + (size_t)ncol * DPAD;

        v8f c0 = {0.f,0.f,0.f,0.f,0.f,0.f,0.f,0.f};
        v8f c1 = c0, c2 = c0, c3 = c0;

        #pragma unroll 2
        for (int k = 0; k < DPAD; k += 4) {
            const int kk = k + koff;                       // even -> 8B aligned
            const v2f b  = *(const v2f*)(wcol + kk);
            const v2f a0 = *(const v2f*)(&sh_eh[m     ][kk]);
            const v2f a1 = *(const v2f*)(&sh_eh[m + 16][kk]);
            const v2f a2 = *(const v2f*)(&sh_eh[m + 32][kk]);
            const v2f a3 = *(const v2f*)(&sh_eh[m + 48][kk]);
            c0 = __builtin_amdgcn_wmma_f32_16x16x4_f32(false, a0, false, b, (short)0, c0, false, false);
            c1 = __builtin_amdgcn_wmma_f32_16x16x4_f32(false, a1, false, b, (short)0, c1, false, false);
            c2 = __builtin_amdgcn_wmma_f32_16x16x4_f32(false, a2, false, b, (short)0, c2, false, false);
            c3 = __builtin_amdgcn_wmma_f32_16x16x4_f32(false, a3, false, b, (short)0, c3, false, false);
        }

        // C layout: VGPR r -> row r (lanes 0-15) / row r+8 (lanes 16-31)
        const int rbase = (lane < 16) ? 0 : 8;
        const int ccol  = wave * 16 + (lane & 15);
        #pragma unroll
        for (int r = 0; r < 8; ++r) {
            sh_em[rbase + r     ][ccol] = c0[r];
            sh_em[rbase + r + 16][ccol] = c1[r];
            sh_em[rbase + r + 32][ccol] = c2[r];
            sh_em[rbase + r + 48][ccol] = c3[r];
        }
    }
    __syncthreads();

    // ---- Phase 3: per-edge attention scalar
    if (tid < TILE_E) {
        float dotv = 0.0f;
        #pragma unroll 4
        for (int j = 0; j < OUT_F; ++j) dotv += sh_em[tid][j] * sh_w2[j];
        const float lr = (dotv > 0.0f) ? dotv : ALPHA * dotv;   // leaky_relu
        float ee = __expf(-lr);
        const int s = sh_src[tid];
        if (s < 0) ee = 0.0f;
        sh_ee[tid] = ee;
        if (s >= 0) atomicAdd(&rowsum[s], ee);
    }
    __syncthreads();

    // ---- Phase 4: scatter-accumulate h[src,:] += ee * em (4 threads/edge)
    const int s2 = sh_src[er];
    if (s2 >= 0) {
        const float ee = sh_ee[er];
        float* __restrict__ hrow = h + (size_t)s2 * OUT_F;
        #pragma unroll 5
        for (int i = 0; i < OUT_F / 4; ++i) {
            const int col = cg * (OUT_F / 4) + i;
            atomicAdd(&hrow[col], ee * sh_em[er][col]);
        }
    }
}

// ---------------------------------------------------------------------------
// Finalize: out = elu(h / rowsum) for both branches (concatenated in d_out).
// ---------------------------------------------------------------------------
__global__ __launch_bounds__(THREADS)
void gat_finalize_kernel(const float* __restrict__ h1, const float* __restrict__ rs1,
                         const float* __restrict__ h2, const float* __restrict__ rs2,
                         float* __restrict__ out)
{
    const int NO  = N_NODES * OUT_F;
    const int gid = blockIdx.x * blockDim.x + threadIdx.x;
    if (gid >= 2 * NO) return;
    const float* hb;
    const float* rb;
    int rem;
    if (gid < NO) { hb = h1; rb = rs1; rem = gid; }
    else          { hb = h2; rb = rs2; rem = gid - NO; }
    const int node = rem / OUT_F;
    float rs = rb[node];
    rs = (rs == 0.0f) ? 1e-12f : rs;
    const float v = hb[rem] / rs;
    out[gid] = (v > 0.0f) ? v : expm1f(v);   // ELU(alpha=1)
}

// ---------------------------------------------------------------------------
extern "C" void kernel_launch(void* const* d_in, const int* in_sizes, int n_in,
                              void* d_out, int out_size, void* d_ws, size_t ws_size,
                              hipStream_t stream) {
    (void)in_sizes; (void)n_in; (void)out_size; (void)ws_size;

    const float* x        = (const float*)d_in[0];   // [N, 100]
    const int*   edge     = (const int*)  d_in[1];   // [2, E]
    const int*   new_edge = (const int*)  d_in[2];   // [2, E]
    const float* eembed   = (const float*)d_in[3];   // [E, 100]
    const float* neembed  = (const float*)d_in[4];   // [E, 100]
    const float* neother  = (const float*)d_in[5];   // [4, E]
    // d_in[6], d_in[7]: empty n-hop tensors, ignored
    const float* a        = (const float*)d_in[8];   // [100, 300]
    const float* new_a    = (const float*)d_in[9];   // [100, 304]
    const float* a2       = (const float*)d_in[10];  // [1, 100]
    const float* new_a2   = (const float*)d_in[11];  // [1, 100]

    float* out = (float*)d_out;

    // workspace layout: h1 | h2 | rs1 | rs2 | Wpad1 | Wpad2
    const size_t NO = (size_t)N_NODES * OUT_F;
    float* h1    = (float*)d_ws;
    float* h2    = h1 + NO;
    float* rs1   = h2 + NO;
    float* rs2   = rs1 + N_NODES;
    float* Wpad1 = rs2 + N_NODES;                    // byte offset 40,400,000 (8B aligned)
    float* Wpad2 = Wpad1 + (size_t)OUTPAD * DPAD;

    // zero only the accumulators (Wpad fully overwritten by prep kernel)
    hipMemsetAsync(d_ws, 0, (2 * NO + 2 * (size_t)N_NODES) * sizeof(float), stream);

    const int PER = OUTPAD * DPAD;
    gat_prep_w_kernel<<<(2 * PER + THREADS - 1) / THREADS, THREADS, 0, stream>>>(
        a, new_a, Wpad1, Wpad2);

    const int eblocks = (N_EDGES + TILE_E - 1) / TILE_E;   // 7813
    gat_edge_kernel<<<eblocks, THREADS, 0, stream>>>(
        x, edge, eembed, /*eother=*/nullptr, Wpad1, a2, /*extra=*/0, h1, rs1);
    gat_edge_kernel<<<eblocks, THREADS, 0, stream>>>(
        x, new_edge, neembed, neother, Wpad2, new_a2, /*extra=*/4, h2, rs2);

    const int total = 2 * (int)NO;
    gat_finalize_kernel<<<(total + THREADS - 1) / THREADS, THREADS, 0, stream>>>(
        h1, rs1, h2, rs2, out);
}